// MonarchEmbedding_14465449853580
// MI455X (gfx1250) — compile-verified
//
#include <hip/hip_runtime.h>

#ifndef __has_builtin
#define __has_builtin(x) 0
#endif

typedef int v4i __attribute__((ext_vector_type(4)));

namespace {
constexpr int kBL1 = 786;   // L: (64, 786, 16)
constexpr int kBL2 = 16;
constexpr int kKR  = 32;    // R: (32, 32, 32)
constexpr int kBR1 = 32;
constexpr int kBR2 = 32;
constexpr int kRow = kKR * kBR2;           // 1024 floats per output row
constexpr int kRElems = kKR * kBR1 * kBR2; // 32768 floats = 128 KB
constexpr int kNTok = 8 * 2048;            // 16384 rows to emit
constexpr int kThreads = 256;              // 8 waves (wave32)
constexpr int kTokPerBlock = 16;
constexpr int kBlocks = kNTok / kTokPerBlock; // 1024
} // namespace

// out[t, kr*32+o] = ((kr&1)==k_hi) ? L[x[t]*16 + (kr>>1)] * R[kr, k_lo, o] : 0
//   with k = x[t]/786, k_hi = k>>5, k_lo = k&31.
// One block emits one 4KB row per loop iter: thread tid stores float4 at col tid*4.
__global__ __launch_bounds__(kThreads) void monarch_embed_rows(
    const int* __restrict__ x,
    const float* __restrict__ L,
    const float* __restrict__ R,
    float* __restrict__ out)
{
    extern __shared__ float sR[];   // 128 KB: full R tensor
    const int tid = threadIdx.x;

    // ---- Stage R into LDS via CDNA5 async global->LDS DMA (ASYNCcnt path) ----
#if __has_builtin(__builtin_amdgcn_global_load_async_to_lds_b128)
    {
        const float* g = R + tid * 4;   // 16 B per lane per issue
        float* l = sR + tid * 4;
#pragma unroll
        for (int i = 0; i < kRElems / (kThreads * 4); ++i) { // 32 issues/thread
            __builtin_amdgcn_global_load_async_to_lds_b128(
                (v4i*)(g + i * kThreads * 4),
                (v4i*)(l + i * kThreads * 4),
                /*offset=*/0, /*cpol=*/0);
        }
    }
#if __has_builtin(__builtin_amdgcn_s_wait_asynccnt)
    __builtin_amdgcn_s_wait_asynccnt(0);
#else
    asm volatile("s_wait_asynccnt 0" ::: "memory");
#endif
#else
    // Fallback (also what the host compilation pass parses): plain vector copy.
    for (int i = tid; i < kRElems / 4; i += kThreads) {
        reinterpret_cast<float4*>(sR)[i] =
            reinterpret_cast<const float4*>(R)[i];
    }
#endif
    __syncthreads();

    // Per-thread constant column decomposition.
    const int kr     = tid >> 3;        // 0..31  (R outer block)
    const int o      = (tid & 7) * 4;   // 0,4,...,28 (float4-aligned inside block)
    const int j      = kr >> 1;         // L column 0..15
    const int parity = kr & 1;
    const int out_col = tid * 4;

    int t = blockIdx.x * kTokPerBlock;
    const int tend = (t + kTokPerBlock < kNTok) ? (t + kTokPerBlock) : kNTok;
    for (; t < tend; ++t) {
        const int idx  = x[t];          // in [0, 50304)
        const int k    = idx / kBL1;    // strength-reduced by compiler
        const int k_hi = k >> 5;
        const int k_lo = k & 31;

        float s = L[idx * kBL2 + j];    // L[k, v, j]; idx == k*786+v
        s = (parity == k_hi) ? s : 0.0f;   // branchless zero block

        const float4 r4 = *reinterpret_cast<const float4*>(
            &sR[(kr * kBR1 + k_lo) * kBR2 + o]);   // ds_load_b128

        float4 res;
        res.x = s * r4.x;
        res.y = s * r4.y;
        res.z = s * r4.z;
        res.w = s * r4.w;
        *reinterpret_cast<float4*>(&out[(size_t)t * kRow + out_col]) = res; // b128 store
    }
}

extern "C" void kernel_launch(void* const* d_in, const int* in_sizes, int n_in,
                              void* d_out, int out_size, void* d_ws, size_t ws_size,
                              hipStream_t stream) {
    const int*   x = (const int*)d_in[0];     // (8, 2048) int32
    const float* L = (const float*)d_in[1];   // (64, 786, 16) f32
    const float* R = (const float*)d_in[2];   // (32, 32, 32) f32
    // d_in[3] (p) is a fixed perfect shuffle, folded analytically into indexing.
    float* out = (float*)d_out;               // (8, 2048, 1024) f32

    const size_t lds_bytes = kRElems * sizeof(float); // 128 KB of 320 KB/WGP
    monarch_embed_rows<<<dim3(kBlocks), dim3(kThreads), lds_bytes, stream>>>(
        x, L, R, out);
}